// AttentionAggregation_13752485282205
// MI455X (gfx1250) — compile-verified
//
#include <hip/hip_runtime.h>
#include <hip/hip_bf16.h>
#include <math.h>

#define N_INST   262144
#define ENC      512         // E
#define HID      256         // H
#define NBAGS    512
#define ROWS     32          // instance rows per tile
#define TPB      4           // tiles per block (double-buffered pipeline)
#define LDSS     516         // padded LDS row stride (floats): row*516 % 64 spreads banks

typedef __attribute__((ext_vector_type(2))) float v2f;
typedef __attribute__((ext_vector_type(8))) float v8f;

// ---------------------------------------------------------------- zero scratch
__global__ void kzero(float* sums) {
    int t = threadIdx.x;
    if (t < NBAGS) sums[t] = 0.0f;
}

// ---- CDNA5 async global->LDS copy (ASYNCcnt-tracked, bypasses VGPRs) --------
__device__ __forceinline__ void async_b128_to_lds(unsigned lds_off, const float* gaddr) {
    asm volatile("global_load_async_to_lds_b128 %0, %1, off"
                 :: "v"(lds_off), "v"(gaddr)
                 : "memory");
}
__device__ __forceinline__ void wait_async0() {
    asm volatile("s_wait_asynccnt 0x0" ::: "memory");
}

// stage one 32 x 512 fp32 tile of X into LDS buffer (16 x b128 per thread)
__device__ __forceinline__ void stage_tile_async(const float* __restrict__ X,
                                                 int row0, float* buf, int tid) {
    unsigned lbase = (unsigned)(size_t)(void*)buf;   // low 32 bits = LDS byte offset
    const float* g0 = X + (size_t)row0 * ENC;
    #pragma unroll
    for (int i = 0; i < 16; ++i) {
        int idx = tid + i * 256;                     // 0..4095
        int r   = idx >> 7;                          // 128 float4 per row
        int c4  = idx & 127;
        unsigned loff = lbase + (unsigned)((r * LDSS + c4 * 4) * 4);
        async_b128_to_lds(loff, g0 + (size_t)r * ENC + c4 * 4);
    }
}

// ------------------------------------------------ attention scores (WMMA GEMM)
// alpha[n] = exp( sum_h tanh( X[n,:]·Vw[h,:] + Vb[h] ) * ww[h] + wb )
// sums[bag[n]] += alpha[n]
__global__ __launch_bounds__(256) void kattn(const float* __restrict__ X,
                                             const float* __restrict__ Vw,
                                             const float* __restrict__ Vb,
                                             const float* __restrict__ ww,
                                             const float* __restrict__ wb,
                                             const int*   __restrict__ bidx,
                                             float* __restrict__ alpha,
                                             float* __restrict__ sums) {
    __shared__ float sh[2][ROWS * LDSS];   // double-buffered 32 x 512 X tiles
    __shared__ float rowsum[ROWS];

    const int tid    = threadIdx.x;
    const int lane   = tid & 31;
    const int wid    = tid >> 5;           // 8 waves
    const int lane16 = lane & 15;
    const int koff   = (lane >> 4) << 1;   // lanes 16..31 hold K=2,3
    const int nbase  = blockIdx.x * (ROWS * TPB);

    // wave's two 16-wide column tiles of H
    const int col0 = (wid * 2 + 0) * 16 + lane16;
    const int col1 = (wid * 2 + 1) * 16 + lane16;
    const float* VwR0 = Vw + (size_t)col0 * ENC;
    const float* VwR1 = Vw + (size_t)col1 * ENC;
    const float wc0 = ww[col0], wc1 = ww[col1];
    const float vb0 = Vb[col0], vb1 = Vb[col1];
    const float wbv = wb[0];
    const int rbase = (lane >> 4) * 8;     // C layout: VGPR i -> rows {i, i+8}

    // prologue: stage tile 0
    stage_tile_async(X, nbase, &sh[0][0], tid);
    wait_async0();
    __syncthreads();

    for (int t = 0; t < TPB; ++t) {
        const int cur = t & 1;
        // prefetch next tile into the other buffer while we compute
        if (t + 1 < TPB)
            stage_tile_async(X, nbase + (t + 1) * ROWS, &sh[cur ^ 1][0], tid);

        if (tid < ROWS) rowsum[tid] = 0.0f;
        __syncthreads();

        const float* shA0 = &sh[cur][lane16 * LDSS];
        const float* shA1 = &sh[cur][(lane16 + 16) * LDSS];

        v8f acc00 = {}; v8f acc01 = {};    // rowtile0 x {col0,col1}
        v8f acc10 = {}; v8f acc11 = {};    // rowtile1 x {col0,col1}

        for (int k0 = 0; k0 < ENC; k0 += 4) {
            const int ka = k0 + koff;
            float2 tb0 = *(const float2*)(VwR0 + ka);
            float2 tb1 = *(const float2*)(VwR1 + ka);
            float2 ta0 = *(const float2*)(shA0 + ka);
            float2 ta1 = *(const float2*)(shA1 + ka);
            v2f a0; a0.x = ta0.x; a0.y = ta0.y;
            v2f a1; a1.x = ta1.x; a1.y = ta1.y;
            v2f b0; b0.x = tb0.x; b0.y = tb0.y;
            v2f b1; b1.x = tb1.x; b1.y = tb1.y;
            acc00 = __builtin_amdgcn_wmma_f32_16x16x4_f32(false, a0, false, b0, (short)0, acc00, false, false);
            acc10 = __builtin_amdgcn_wmma_f32_16x16x4_f32(false, a1, false, b0, (short)0, acc10, false, false);
            acc01 = __builtin_amdgcn_wmma_f32_16x16x4_f32(false, a0, false, b1, (short)0, acc01, false, false);
            acc11 = __builtin_amdgcn_wmma_f32_16x16x4_f32(false, a1, false, b1, (short)0, acc11, false, false);
        }

        // epilogue: tanh(+Vb), dot with ww, per-row reduce in LDS
        #pragma unroll
        for (int i = 0; i < 8; ++i) {
            int r = rbase + i;
            float p0 = tanhf(acc00[i] + vb0) * wc0 + tanhf(acc01[i] + vb1) * wc1;
            atomicAdd(&rowsum[r], p0);
            float p1 = tanhf(acc10[i] + vb0) * wc0 + tanhf(acc11[i] + vb1) * wc1;
            atomicAdd(&rowsum[r + 16], p1);
        }
        __syncthreads();

        if (tid < ROWS) {
            int n = nbase + t * ROWS + tid;
            float a = expf(rowsum[tid] + wbv);
            alpha[n] = a;
            atomicAdd(&sums[bidx[n]], a);
        }

        // next buffer must be resident; also fences rowsum readers before re-zero
        wait_async0();
        __syncthreads();
    }
}

// ------------------------------- per-bag weighted pooling + logits + softmax
__device__ __forceinline__ int lower_bound_i(const int* a, int n, int key) {
    int lo = 0, hi = n;
    while (lo < hi) { int m = (lo + hi) >> 1; if (a[m] < key) lo = m + 1; else hi = m; }
    return lo;
}

__global__ __launch_bounds__(256) void kpool(const float* __restrict__ X,
                                             const float* __restrict__ dw,
                                             const float* __restrict__ db,
                                             const int*   __restrict__ bidx,
                                             const float* __restrict__ alpha,
                                             const float* __restrict__ sums,
                                             float* __restrict__ out) {
    __shared__ int s_lo, s_hi;
    __shared__ float red0[256], red1[256];

    const int b   = blockIdx.x;
    const int tid = threadIdx.x;
    if (tid == 0) {
        s_lo = lower_bound_i(bidx, N_INST, b);
        s_hi = lower_bound_i(bidx, N_INST, b + 1);
    }
    __syncthreads();
    const int lo = s_lo, hi = s_hi;
    const float invS = (lo < hi) ? (1.0f / sums[b]) : 0.0f;

    // each thread owns 2 encoding columns (512 / 256)
    const float2* X2 = (const float2*)X;   // [N, 256] float2, col pair = tid
    float acc0 = 0.0f, acc1 = 0.0f;
    for (int n = lo; n < hi; ++n) {
        float w = alpha[n] * invS;
        float2 x = X2[(size_t)n * 256 + tid];
        acc0 += w * x.x;
        acc1 += w * x.y;
    }

    // fused logits: dw is [2, E] row-major
    const int e0 = tid * 2, e1 = tid * 2 + 1;
    red0[tid] = acc0 * dw[e0]       + acc1 * dw[e1];
    red1[tid] = acc0 * dw[ENC + e0] + acc1 * dw[ENC + e1];
    __syncthreads();
    for (int s = 128; s > 0; s >>= 1) {
        if (tid < s) { red0[tid] += red0[tid + s]; red1[tid] += red1[tid + s]; }
        __syncthreads();
    }
    if (tid == 0) {
        float l0 = red0[0] + db[0];
        float l1 = red1[0] + db[1];
        float m  = fmaxf(l0, l1);
        float z0 = expf(l0 - m), z1 = expf(l1 - m);
        float inv = 1.0f / (z0 + z1);
        out[b * 2 + 0] = z0 * inv;
        out[b * 2 + 1] = z1 * inv;
    }
}

extern "C" void kernel_launch(void* const* d_in, const int* in_sizes, int n_in,
                              void* d_out, int out_size, void* d_ws, size_t ws_size,
                              hipStream_t stream) {
    const float* X    = (const float*)d_in[0];   // [N, E]
    const float* Vw   = (const float*)d_in[1];   // [H, E]
    const float* Vb   = (const float*)d_in[2];   // [H]
    const float* ww   = (const float*)d_in[3];   // [1, H]
    const float* wb   = (const float*)d_in[4];   // [1]
    const float* dw   = (const float*)d_in[5];   // [2, E]
    const float* db   = (const float*)d_in[6];   // [2]
    const int*   bi   = (const int*)  d_in[7];   // [N]
    float*       out  = (float*)d_out;           // [NBAGS, 2]

    float* ws_alpha = (float*)d_ws;              // N floats
    float* ws_sums  = ws_alpha + N_INST;         // NBAGS floats

    kzero<<<1, NBAGS, 0, stream>>>(ws_sums);
    kattn<<<N_INST / (ROWS * TPB), 256, 0, stream>>>(X, Vw, Vb, ww, wb, bi, ws_alpha, ws_sums);
    kpool<<<NBAGS, 256, 0, stream>>>(X, dw, db, bi, ws_alpha, ws_sums, out);

    (void)in_sizes; (void)n_in; (void)out_size; (void)ws_size;
}